// CtcLoss_80410377716050
// MI455X (gfx1250) — compile-verified
//
#include <hip/hip_runtime.h>
#include <stdint.h>

// Compact emission row: [0..S-1] = label probs, [S]=blank prob, padded to 136
// floats (544 B = 34 x 16 B, keeps float4 LDS reads aligned).
#define SROW 136
#define CH   24   // time rows per TDM chunk (4*24*136*4 = 52,224 B LDS ring)
#define NSLOT 4
#define PF    3   // prefetch depth (chunks in flight)

typedef unsigned int u32x4 __attribute__((ext_vector_type(4)));
typedef int          i32x8 __attribute__((ext_vector_type(8)));
typedef int          i32x4 __attribute__((ext_vector_type(4)));

// Issue one Tensor-Data-Mover load of `rows` consecutive 136-float rows from
// global memory into LDS at byte offset lds_off. D# per CDNA5 ISA ch.8.
__device__ __forceinline__ void tdm_load_rows(const float* gsrc, uint32_t lds_off, int rows) {
  uint64_t ga = (uint64_t)(uintptr_t)gsrc;
  u32x4 g0;
  g0.x = 1u;                                               // count=1 (valid D#)
  g0.y = lds_off;                                          // lds_addr
  g0.z = (uint32_t)ga;                                     // global_addr[31:0]
  g0.w = (uint32_t)((ga >> 32) & 0x01FFFFFFull) | (2u << 30); // addr[56:32] | type=2
  i32x8 g1;
  g1[0] = (int)(2u << 16);                 // data_size=2 (4 bytes); no multicast
  g1[1] = (int)((unsigned)SROW << 16);     // tensor_dim0[15:0]
  g1[2] = (int)((unsigned)rows << 16);     // tensor_dim0[31:16]=0 | tensor_dim1[15:0]
  g1[3] = (int)((unsigned)SROW << 16);     // tensor_dim1[31:16]=0 | tile_dim0
  g1[4] = rows;                            // tile_dim1 | tile_dim2=0
  g1[5] = SROW;                            // tensor_dim0_stride[31:0]
  g1[6] = 0;                               // stride hi | dim1_stride lo
  g1[7] = 0;
  i32x4 z4 = {0, 0, 0, 0};                 // groups 2/3 unused (2-D tile)
  i32x8 z8 = {0, 0, 0, 0, 0, 0, 0, 0};     // extra group (6-arg builtin form)
  __builtin_amdgcn_tensor_load_to_lds(g0, g1, z4, z4, z8, 0);
}

// Phase 1: full-GPU gather + exp. One block per (b,t) row.
__global__ __launch_bounds__(160) void ctc_gather_kernel(
    const float* __restrict__ lp, const int* __restrict__ targets,
    float* __restrict__ E, int T, int V, int S) {
  const int row = blockIdx.x;            // b*T + t
  const int j   = threadIdx.x;
  const int b   = row / T;
  const float* src = lp + (size_t)row * V;
  if (j < S) {
    E[(size_t)row * SROW + j] = __expf(src[targets[b * S + j]]);
  } else if (j == S) {
    E[(size_t)row * SROW + S] = __expf(src[0]);   // blank
  }
}

// Phase 2: one wave32 per sequence. Lane k owns states 8k..8k+7; lane 31 also
// owns state 256. Scaled-probability forward DP, TDM-prefetched emissions.
__global__ __launch_bounds__(32) void ctc_dp_kernel(
    const float* __restrict__ E, const int* __restrict__ targets,
    const int* __restrict__ ilen, const int* __restrict__ tlen,
    float* __restrict__ lossb, int T, int S) {
  __shared__ __align__(16) float ring[NSLOT][CH * SROW];
  const int b    = blockIdx.x;
  const int lane = threadIdx.x;
  const int Tact = ilen[b];
  const int tl   = tlen[b];
  if (Tact <= 0) { if (lane == 0) lossb[b] = 0.f; return; }
  const float* Eb = E + (size_t)b * T * SROW;
  const int nchunk = (Tact + CH - 1) / CH;

  // Per-lane skip masks: odd state l=8k+2j+1 (label s=4k+j) may skip from l-2
  // iff s>0 and targets[s] != targets[s-1] (labels are nonzero by construction).
  const int sbase = b * S + 4 * lane;
  const int tp = (lane == 0) ? -2147483647 : targets[sbase - 1];
  const int t0 = targets[sbase + 0], t1 = targets[sbase + 1];
  const int t2 = targets[sbase + 2], t3 = targets[sbase + 3];
  const float sm0 = (lane > 0 && t0 != tp) ? 1.f : 0.f;
  const float sm1 = (t1 != t0) ? 1.f : 0.f;
  const float sm2 = (t2 != t1) ? 1.f : 0.f;
  const float sm3 = (t3 != t2) ? 1.f : 0.f;

  // Prime the TDM pipeline (clamped chunk indices keep exactly PF in flight).
  for (int i = 0; i < PF; ++i) {
    int cc = (i < nchunk) ? i : (nchunk - 1);
    int rows = min(CH, Tact - cc * CH);
    tdm_load_rows(Eb + (size_t)cc * CH * SROW, (uint32_t)(uintptr_t)&ring[i][0], rows);
  }

  float a0=0.f,a1=0.f,a2=0.f,a3=0.f,a4=0.f,a5=0.f,a6=0.f,a7=0.f,a8=0.f;
  double lscale = 0.0;

  for (int c = 0; c < nchunk; ++c) {
    {   // issue chunk c+PF into the free ring slot (clamped re-issue at tail)
      int cc = c + PF; if (cc > nchunk - 1) cc = nchunk - 1;
      int rows = min(CH, Tact - cc * CH);
      tdm_load_rows(Eb + (size_t)cc * CH * SROW,
                    (uint32_t)(uintptr_t)&ring[(c + PF) & (NSLOT - 1)][0], rows);
    }
    __builtin_amdgcn_s_wait_tensorcnt(PF);   // chunk c (oldest of PF+1) is done
    asm volatile("" ::: "memory");
    const float* slotp = &ring[c & (NSLOT - 1)][0];
    const int rows = min(CH, Tact - c * CH);
    for (int r = 0; r < rows; ++r) {
      const int t = c * CH + r;
      const float* row = slotp + r * SROW;
      const float  pb  = row[128];                      // blank prob, broadcast
      const float4 p   = ((const float4*)row)[lane];    // labels 4k..4k+3
      if (t == 0) {
        if (lane == 0) { a0 = pb; a1 = p.x; }           // alpha0 init
      } else {
        float up7 = __shfl_up(a7, 1, 32);               // old alpha[8k-1]
        if (lane == 0) up7 = 0.f;
        const float n0 = (a0 + up7) * pb;               // l=8k   (blank)
        const float n1 = (a1 + a0 + sm0 * up7) * p.x;   // l=8k+1
        const float n2 = (a2 + a1) * pb;                // l=8k+2 (blank)
        const float n3 = (a3 + a2 + sm1 * a1) * p.y;    // l=8k+3
        const float n4 = (a4 + a3) * pb;                // l=8k+4 (blank)
        const float n5 = (a5 + a4 + sm2 * a3) * p.z;    // l=8k+5
        const float n6 = (a6 + a5) * pb;                // l=8k+6 (blank)
        const float n7 = (a7 + a6 + sm3 * a5) * p.w;    // l=8k+7
        const float n8 = (a8 + a7) * pb;                // l=256 (lane 31 only)
        a0=n0; a1=n1; a2=n2; a3=n3; a4=n4; a5=n5; a6=n6; a7=n7; a8=n8;
      }
      if ((t & 3) == 3) {                                // renormalize
        float m = fmaxf(fmaxf(fmaxf(fmaxf(a0,a1), fmaxf(a2,a3)),
                              fmaxf(fmaxf(a4,a5), fmaxf(a6,a7))), a8);
        for (int off = 16; off > 0; off >>= 1)
          m = fmaxf(m, __shfl_xor(m, off, 32));
        const float inv = 1.0f / m;
        a0*=inv; a1*=inv; a2*=inv; a3*=inv; a4*=inv;
        a5*=inv; a6*=inv; a7*=inv; a8*=inv;
        lscale += (double)__logf(m);
      }
    }
  }

  // Drain outstanding TDM writes before reusing slot 0 as scratch.
  __builtin_amdgcn_s_wait_tensorcnt(0);
  asm volatile("" ::: "memory");
  float* fin = &ring[0][0];
  __syncthreads();
  fin[8*lane+0]=a0; fin[8*lane+1]=a1; fin[8*lane+2]=a2; fin[8*lane+3]=a3;
  fin[8*lane+4]=a4; fin[8*lane+5]=a5; fin[8*lane+6]=a6; fin[8*lane+7]=a7;
  if (lane == 31) fin[256] = a8;
  __syncthreads();
  if (lane == 0) {
    int idx1 = 2 * tl;
    int idx2 = idx1 - 1; if (idx2 < 0) idx2 = 0;
    const float s = fin[idx1] + fin[idx2];
    const double ll = (double)__logf(s) + lscale;
    lossb[b] = (float)(-ll);
  }
}

// Phase 3: sum per-sequence losses into the scalar output.
__global__ __launch_bounds__(32) void ctc_reduce_kernel(
    const float* __restrict__ lossb, float* __restrict__ out, int B) {
  float v = 0.f;
  for (int i = threadIdx.x; i < B; i += 32) v += lossb[i];
  for (int off = 16; off > 0; off >>= 1) v += __shfl_down(v, off, 32);
  if (threadIdx.x == 0) out[0] = v;
}

extern "C" void kernel_launch(void* const* d_in, const int* in_sizes, int n_in,
                              void* d_out, int out_size, void* d_ws, size_t ws_size,
                              hipStream_t stream) {
  const float* lp      = (const float*)d_in[0];
  const int*   targets = (const int*)d_in[1];
  const int*   ilen    = (const int*)d_in[2];
  const int*   tlen    = (const int*)d_in[3];
  const int B = in_sizes[2];
  const int S = in_sizes[1] / B;                    // 128
  const int V = 1024;                               // fixed by problem instance
  const int T = (int)(in_sizes[0] / ((long long)B * V));  // 2000

  float* E     = (float*)d_ws;                      // [B][T][SROW] = 34.8 MB
  float* lossb = E + (size_t)B * T * SROW;          // [B]

  ctc_gather_kernel<<<B * T, 160, 0, stream>>>(lp, targets, E, T, V, S);
  ctc_dp_kernel<<<B, 32, 0, stream>>>(E, targets, ilen, tlen, lossb, T, S);
  ctc_reduce_kernel<<<1, 32, 0, stream>>>(lossb, (float*)d_out, B);
}